// NDCGLoss_8564164788534
// MI455X (gfx1250) — compile-verified
//
#include <hip/hip_runtime.h>
#include <hip/hip_bf16.h>

// ---------------------------------------------------------------------------
// NDCG / LambdaRank loss for MI455X (gfx1250, wave32).
//
// Kernel 1: one workgroup (128 threads = 4 waves) per query.
//   - gains via exp2, grade histogram replaces the sort for ideal DCG
//   - pairwise |dg| * log2(sigmoid(ds)) loop, j-operands broadcast from LDS
//   - per-wave reduction via V_WMMA_F32_16X16X4_F32 (A = ones) + shfl_xor
// Kernel 2: deterministic reduction of B per-query values -> d_out[0].
// ---------------------------------------------------------------------------

typedef __attribute__((ext_vector_type(2))) float v2f;
typedef __attribute__((ext_vector_type(8))) float v8f;

#define LOG2E 1.4426950408889634f

// Reduce one value per lane across the 32-lane wave; every lane returns the sum.
// Uses WMMA f32 16x16x4 with A = ones: D[m][n] = sum_k B[k][n], so with the
// lane partial in B-VGPR0 and 0 in B-VGPR1, colsum[n] = p_n + p_{16+n}
// independent of which k-row each 16-lane half maps to. D VGPR0 then carries
// colsum[n mod 16] in every lane; 4 xor-shuffles finish the 16-way sum.
__device__ __forceinline__ float wave_reduce_wmma(float v) {
    v2f a; a.x = 1.0f; a.y = 1.0f;         // 16x4 ones
    v2f b; b.x = v;    b.y = 0.0f;         // 4x16 fragment: partials + zeros
    v8f c = {};
    v8f d = __builtin_amdgcn_wmma_f32_16x16x4_f32(
        /*neg_a=*/false, a, /*neg_b=*/false, b,
        /*c_mod=*/(short)0, c, /*reuse_a=*/false, /*reuse_b=*/false);
    float x = d.s0;                         // colsum[lane mod 16]
    x += __shfl_xor(x, 1, 32);
    x += __shfl_xor(x, 2, 32);
    x += __shfl_xor(x, 4, 32);
    x += __shfl_xor(x, 8, 32);
    return x;                               // full 32-lane sum, all lanes
}

__global__ void __launch_bounds__(128)
ndcg_per_query_kernel(const float* __restrict__ scores,
                      const int*  __restrict__ relevances,
                      const int*  __restrict__ qlen,
                      float* __restrict__ per_query,
                      int D) {
    const int b   = blockIdx.x;
    const int tid = threadIdx.x;

    __shared__ float s_s[128];
    __shared__ float s_g[128];
    __shared__ float s_wsum[4];
    __shared__ float s_idcg;
    __shared__ int   s_cnt[5];

    if (tid < 5) s_cnt[tid] = 0;

    const int   L     = qlen[b];
    const float sc    = scores[(long)b * D + tid];
    const int   rv    = relevances[(long)b * D + tid];
    const bool  valid = (tid < L);
    const float g     = valid ? (exp2f((float)rv) - 1.0f) : 0.0f;

    s_s[tid] = sc;
    s_g[tid] = g;
    __syncthreads();

    if (valid && rv >= 0 && rv < 5) atomicAdd(&s_cnt[rv], 1);
    __syncthreads();

    // Ideal DCG: grades are integers 0..4, so descending sort == histogram walk.
    if (tid == 0) {
        float idcg = 0.0f;
        int rank = 0;
        for (int gr = 4; gr >= 1; --gr) {
            const float gain = exp2f((float)gr) - 1.0f;  // 15, 7, 3, 1
            const int   c    = s_cnt[gr];
            for (int k = 0; k < c; ++k) {
                // discount = 1 / log2(rank + 2)
                idcg += gain / __builtin_amdgcn_logf((float)(rank + 2));
                ++rank;
            }
        }
        s_idcg = idcg;
    }

    // Pairwise: acc_i = sum_{j<L} |g_i - g_j| * log2(sigmoid(s_i - s_j))
    // log2sigmoid(x) = min(x,0)*log2(e) - log2(1 + 2^(-|x|*log2(e)))
    float acc = 0.0f;
    if (valid) {
        for (int j = 0; j < L; ++j) {
            const float dd    = sc - s_s[j];
            const float delta = fabsf(g - s_g[j]);
            const float ax    = fabsf(dd);
            const float t     = __builtin_amdgcn_exp2f(-ax * LOG2E);
            const float l2s   = fminf(dd, 0.0f) * LOG2E
                              - __builtin_amdgcn_logf(1.0f + t);
            acc += delta * l2s;
        }
    }

    // Full-EXEC point: per-wave WMMA reduction (EXEC must be all 1s here).
    const float wsum = wave_reduce_wmma(acc);
    if ((tid & 31) == 0) s_wsum[tid >> 5] = wsum;
    __syncthreads();

    if (tid == 0) {
        const float num = s_wsum[0] + s_wsum[1] + s_wsum[2] + s_wsum[3];
        per_query[b] = -num / (s_idcg + 1e-8f);
    }
}

// Deterministic final reduction: 1 wave, fixed per-lane serial order, then
// one WMMA wave reduction. No float atomics -> bitwise-stable across replays.
__global__ void __launch_bounds__(32)
ndcg_reduce_kernel(const float* __restrict__ per_query,
                   float* __restrict__ out, int n) {
    const int lane = threadIdx.x;
    float s = 0.0f;
    for (int k = lane; k < n; k += 32) s += per_query[k];
    const float tot = wave_reduce_wmma(s);
    if (lane == 0) out[0] = tot / (float)n;
}

extern "C" void kernel_launch(void* const* d_in, const int* in_sizes, int n_in,
                              void* d_out, int out_size, void* d_ws, size_t ws_size,
                              hipStream_t stream) {
    const float* scores     = (const float*)d_in[0];
    const int*   relevances = (const int*)d_in[1];
    const int*   qlen       = (const int*)d_in[2];

    const int B = in_sizes[2];          // 2048
    const int D = in_sizes[0] / B;      // 128

    float* ws  = (float*)d_ws;          // B floats of per-query loss
    float* out = (float*)d_out;

    ndcg_per_query_kernel<<<B, D, 0, stream>>>(scores, relevances, qlen, ws, D);
    ndcg_reduce_kernel<<<1, 32, 0, stream>>>(ws, out, B);
}